// GaugeField_4569845203311
// MI455X (gfx1250) — compile-verified
//
#include <hip/hip_runtime.h>
#include <hip/hip_bf16.h>
#include <math.h>

typedef __attribute__((ext_vector_type(2))) float v2f;
typedef __attribute__((ext_vector_type(8))) float v8f;

// ---------------------------------------------------------------- zero init
__global__ void gauge_zero_kernel(float* __restrict__ p, long long n) {
    long long i = (long long)blockIdx.x * blockDim.x + threadIdx.x;
    if (i < n) p[i] = 0.0f;
}

// ---------------------------------------------------------------- edge pass
// omega = 0.5*(P - P^T); A[start] += omega*proj_s ; A[end] += -omega*proj_e
// proj_s = <v_s, d>/|d|^2 ; proj_e = <v_e, -d>/|d|^2  (so -omega*proj_e = omega*<v_e,d>/|d|^2)
__global__ void gauge_edge_kernel(const int* __restrict__ edges,
                                  const float* __restrict__ omega_params,
                                  const float* __restrict__ x,
                                  const float* __restrict__ v,
                                  float* __restrict__ A, int E) {
    int e = blockIdx.x * blockDim.x + threadIdx.x;
    if (e >= E) return;
    int s = edges[2 * e];
    int t = edges[2 * e + 1];
    const float* P = omega_params + (long long)e * 9;
    float o01 = 0.5f * (P[1] - P[3]);
    float o02 = 0.5f * (P[2] - P[6]);
    float o12 = 0.5f * (P[5] - P[7]);
    float dx = x[3 * t] - x[3 * s];
    float dy = x[3 * t + 1] - x[3 * s + 1];
    float dz = x[3 * t + 2] - x[3 * s + 2];
    float dq = fmaxf(dx * dx + dy * dy + dz * dz, 1e-6f);
    float ps = (v[3 * s] * dx + v[3 * s + 1] * dy + v[3 * s + 2] * dz) / dq;
    float pe = (v[3 * t] * dx + v[3 * t + 1] * dy + v[3 * t + 2] * dz) / dq;
    float* As = A + (long long)s * 9;
    float* At = A + (long long)t * 9;
    // omega has zero diagonal -> only 6 elements per target
    atomicAdd(As + 1,  o01 * ps); atomicAdd(As + 2,  o02 * ps);
    atomicAdd(As + 3, -o01 * ps); atomicAdd(As + 5,  o12 * ps);
    atomicAdd(As + 6, -o02 * ps); atomicAdd(As + 7, -o12 * ps);
    atomicAdd(At + 1,  o01 * pe); atomicAdd(At + 2,  o02 * pe);
    atomicAdd(At + 3, -o01 * pe); atomicAdd(At + 5,  o12 * pe);
    atomicAdd(At + 6, -o02 * pe); atomicAdd(At + 7, -o12 * pe);
}

// ---------------------------------------------------------------- triangle WMMA pass
// One wave (32 lanes) handles 5 triangles. Block-diagonal packing of 5 3x3
// matrices into the 16x4 A / 4x16 B operands of V_WMMA_F32_16X16X4_F32:
// C[3a+r,3b+c] = (La . Rb)[r,c]; keep diagonal blocks (a==b), discard rest.
// All guarded LDS accesses are made branchless by routing invalid lanes to a
// zero-filled pad slot (loads) and by dumping the full 16x16 C to LDS (stores).
#define UPAD 135   // ldsU zero pad index
#define CPAD 256   // ldsC zero pad index

__global__ void __launch_bounds__(32)
gauge_tri_wmma_kernel(const float* __restrict__ omega_params,
                      const int* __restrict__ tei,
                      const int* __restrict__ signs,
                      float* __restrict__ Omega_out, int T) {
    __shared__ float ldsU[136];   // [slot(3)][tri(5)][3x3] + zero pad
    __shared__ float ldsC[257];   // full 16x16 C scratch + zero pad
    const int L = threadIdx.x;
    const int base = blockIdx.x * 5;

    if (L == 0) { ldsU[UPAD] = 0.0f; ldsC[CPAD] = 0.0f; }

    // Rodrigues expm for 15 (triangle, edge-slot) pairs on lanes 0..14.
    // exp(S)^T = exp(-S) -> sign<0 handled by flipping 'a'.
    if (L < 15) {
        int t = L / 3, w = L % 3;
        int tg = base + t;
        float U[9];
        if (tg < T) {
            int eidx = tei[tg * 3 + w];
            int sg = signs[tg * 3 + w];
            const float* P = omega_params + (long long)eidx * 9;
            float w0 = 0.5f * (P[7] - P[5]);
            float w1 = 0.5f * (P[2] - P[6]);
            float w2 = 0.5f * (P[3] - P[1]);
            float th2 = w0 * w0 + w1 * w1 + w2 * w2;
            float th2c = fmaxf(th2, 1e-24f);
            float th = sqrtf(th2c);
            float a = sinf(th) / th;
            float b = (1.0f - cosf(th)) / th2c;
            if (sg < 0) a = -a;
            float dg = 1.0f - b * th2;  // S*S = w w^T - th2 * I
            U[0] = dg + b * w0 * w0;       U[1] = -a * w2 + b * w0 * w1;  U[2] =  a * w1 + b * w0 * w2;
            U[3] =  a * w2 + b * w0 * w1;  U[4] = dg + b * w1 * w1;       U[5] = -a * w0 + b * w1 * w2;
            U[6] = -a * w1 + b * w0 * w2;  U[7] =  a * w0 + b * w1 * w2;  U[8] = dg + b * w2 * w2;
        } else {
            #pragma unroll
            for (int q = 0; q < 9; ++q) U[q] = 0.0f;
        }
        #pragma unroll
        for (int q = 0; q < 9; ++q) ldsU[w * 45 + t * 9 + q] = U[q];
    }
    __syncthreads();

    const int half = L >> 4;       // lane>=16 half
    const int lm = L & 15;         // A: row m / B,C: col n
    const int d3 = lm / 3;         // block index of this row/col
    const int r3 = lm - 3 * d3;    // position within block
    const bool in15 = (lm < 15);
    v8f cz = {};

    // WMMA #1: A = blockdiag(U2) (slot 1), B = blockdiag(U1) (slot 0)
    v2f af, bf;
    #pragma unroll
    for (int vv = 0; vv < 2; ++vv) {
        int k = vv + 2 * half;
        bool valid = in15 && (k < 3);
        int ia = valid ? (45 + d3 * 9 + r3 * 3 + k) : UPAD;  // U2[row, k]
        int ib = valid ? ( 0 + d3 * 9 + k * 3 + r3) : UPAD;  // U1[k, col]
        af[vv] = ldsU[ia];
        bf[vv] = ldsU[ib];
    }
    v8f d1 = __builtin_amdgcn_wmma_f32_16x16x4_f32(false, af, false, bf,
                                                   (short)0, cz, false, false);
    // dump full C1 to LDS: vgpr r -> row m = r + 8*half, lane -> col n
    {
        const int cb = (8 * half) * 16 + lm;
        #pragma unroll
        for (int r = 0; r < 8; ++r) ldsC[cb + r * 16] = d1[r];
    }
    __syncthreads();

    // WMMA #2: A = blockdiag(U3) (slot 2), B = blockdiag(M), M[k,c] = C1[3*b+k, 3*b+c]
    v2f af2, bf2;
    #pragma unroll
    for (int vv = 0; vv < 2; ++vv) {
        int k = vv + 2 * half;
        bool valid = in15 && (k < 3);
        int ia = valid ? (90 + d3 * 9 + r3 * 3 + k) : UPAD;     // U3[row, k]
        int ib = valid ? ((3 * d3 + k) * 16 + lm)  : CPAD;      // C1[3*d3+k, n]
        af2[vv] = ldsU[ia];
        bf2[vv] = ldsC[ib];
    }
    __syncthreads();  // all B2 reads done before C scratch is overwritten
    v8f d2 = __builtin_amdgcn_wmma_f32_16x16x4_f32(false, af2, false, bf2,
                                                   (short)0, cz, false, false);
    {
        const int cb = (8 * half) * 16 + lm;
        #pragma unroll
        for (int r = 0; r < 8; ++r) ldsC[cb + r * 16] = d2[r];
    }
    __syncthreads();

    // Omega = 0.5 * (H - H^T) with H[i,j] = C2[3t+i, 3t+j]; 45 outputs per wave
    int nvalid = T - base; if (nvalid > 5) nvalid = 5;
    #pragma unroll
    for (int pass = 0; pass < 2; ++pass) {
        int idx = L + pass * 32;
        int t = idx / 9;
        if (idx < 45 && t < nvalid) {
            int e = idx - t * 9, i = e / 3, j = e - 3 * (e / 3);
            float hij = ldsC[(3 * t + i) * 16 + (3 * t + j)];
            float hji = ldsC[(3 * t + j) * 16 + (3 * t + i)];
            Omega_out[(long long)(base + t) * 9 + e] = 0.5f * (hij - hji);
        }
    }
}

// ---------------------------------------------------------------- force pass
__global__ void gauge_force_kernel(const int* __restrict__ tris,
                                   const float* __restrict__ Omega,
                                   const float* __restrict__ x,
                                   const float* __restrict__ v,
                                   const float* __restrict__ J,
                                   float* __restrict__ F, int T) {
    int t = blockIdx.x * blockDim.x + threadIdx.x;
    if (t >= T) return;
    int i = tris[3 * t], j = tris[3 * t + 1], k = tris[3 * t + 2];
    float xi0 = x[3 * i], xi1 = x[3 * i + 1], xi2 = x[3 * i + 2];
    float u0 = x[3 * j] - xi0, u1 = x[3 * j + 1] - xi1, u2 = x[3 * j + 2] - xi2;
    float w0 = x[3 * k] - xi0, w1 = x[3 * k + 1] - xi1, w2 = x[3 * k + 2] - xi2;
    float vi0 = v[3 * i], vi1 = v[3 * i + 1], vi2 = v[3 * i + 2];
    float vw = -vi0 * w0 + vi1 * w1 + vi2 * w2;   // Minkowski(-,+,+)
    float vu = -vi0 * u0 + vi1 * u1 + vi2 * u2;
    float L0 = vw * u0 - vu * w0;
    float L1 = vw * u1 - vu * w1;
    float L2 = vw * u2 - vu * w2;
    const float* Om = Omega + (long long)t * 9;
    // Ovec = (Omega[2,1], Omega[0,2], Omega[1,0]) = elements 7, 2, 3
    float q = J[3 * i] * Om[7] + J[3 * i + 1] * Om[2] + J[3 * i + 2] * Om[3];
    atomicAdd(F + 3 * i,     q * L0);
    atomicAdd(F + 3 * i + 1, q * L1);
    atomicAdd(F + 3 * i + 2, q * L2);
}

// ---------------------------------------------------------------- launch
extern "C" void kernel_launch(void* const* d_in, const int* in_sizes, int n_in,
                              void* d_out, int out_size, void* d_ws, size_t ws_size,
                              hipStream_t stream) {
    (void)n_in; (void)d_ws; (void)ws_size; (void)out_size;
    const int*   edges        = (const int*)d_in[0];    // (E,2)
    const int*   triangles    = (const int*)d_in[1];    // (T,3)
    const int*   tri_edge_idx = (const int*)d_in[2];    // (T,3)
    const int*   tri_signs    = (const int*)d_in[3];    // (T,3)
    const float* omega_params = (const float*)d_in[4];  // (E,3,3)
    const float* x            = (const float*)d_in[5];  // (N,3)
    const float* v            = (const float*)d_in[6];  // (N,3)
    const float* J            = (const float*)d_in[7];  // (N,3)

    const int E = in_sizes[0] / 2;
    const int T = in_sizes[1] / 3;
    const int N = in_sizes[5] / 3;

    float* Omega = (float*)d_out;                              // T*9
    float* A     = Omega + (long long)T * 9;                   // N*9
    float* F     = A + (long long)N * 9;                       // N*3

    // zero A and F (d_out is poisoned by the harness)
    long long nz = (long long)N * 12;
    gauge_zero_kernel<<<(unsigned)((nz + 255) / 256), 256, 0, stream>>>(A, nz);

    gauge_edge_kernel<<<(E + 255) / 256, 256, 0, stream>>>(
        edges, omega_params, x, v, A, E);

    int triBlocks = (T + 4) / 5;  // 5 triangles per wave (2 WMMAs per wave)
    gauge_tri_wmma_kernel<<<triBlocks, 32, 0, stream>>>(
        omega_params, tri_edge_idx, tri_signs, Omega, T);

    gauge_force_kernel<<<(T + 255) / 256, 256, 0, stream>>>(
        triangles, Omega, x, v, J, F, T);
}